// SlotAttention_35905926595058
// MI455X (gfx1250) — compile-verified
//
#include <hip/hip_runtime.h>
#include <hip/hip_bf16.h>

// ---------------------------------------------------------------------------
// SlotAttention (D=768, H=3072, N_SLOTS=128, ITERS=3), fp32, MI455X / gfx1250.
//
// Key identity: softmax over j followed by sum over j against j-independent v
// gives updates == v exactly. So q/k/pano/dots/softmax never touch the output,
// and gi = v@W_ih^T+b_ih is loop-invariant. Remaining work: 5 skinny fp32
// GEMMs (M=128) per iteration schedule + fused GRU/LayerNorm elementwise.
// GEMMs use V_WMMA_F32_16X16X4_F32 (full fp32 precision, matches reference).
// ---------------------------------------------------------------------------

#define D_DIM   768
#define H_DIM   3072
#define NSLOTS  128
#define THREE_D 2304
#define LN_EPS  1e-5f

typedef __attribute__((ext_vector_type(2))) float v2f;
typedef __attribute__((ext_vector_type(8))) float v8f;

// Y[M,N] = X[M,K] @ W[N,K]^T + bias[N] (+ addsrc[M,N]) (optional relu).
// M == 128. Grid: N/16 blocks x 256 threads (8 waves). Wave w in a block
// computes output tile rows [16w, 16w+16), cols [16*blockIdx.x, +16).
//
// fp32 WMMA fragment layout (wave32, ISA 7.12.2):
//   A 16x4 : lane l holds A[l&15, kk+2*(l>>4) .. +1] in 2 VGPRs
//   B 4x16 : lane l holds B[2*(l>>4)..+1, l&15]      in 2 VGPRs
//   With B[k,n] = W[n0+n, kk+k], both operands are the same addressing
//   pattern: float2 at  base + (row0 + (l&15))*K + kk + 2*(l>>4).
//   C/D 16x16: VGPR r, lane l -> element (r + 8*(l>>4), l&15).
__global__ __launch_bounds__(256) void gemm_xwt_wmma(
    const float* __restrict__ X, const float* __restrict__ W,
    const float* __restrict__ bias, const float* __restrict__ addsrc,
    float* __restrict__ Y, int K, int N, int relu)
{
    const int lane = threadIdx.x & 31;
    const int wave = threadIdx.x >> 5;
    const int m0   = wave * 16;
    const int n0   = blockIdx.x * 16;
    const int half = lane >> 4;   // 0: K pair (0,1)   1: K pair (2,3)
    const int l15  = lane & 15;

    const float* __restrict__ xrow = X + (size_t)(m0 + l15) * K + 2 * half;
    const float* __restrict__ wrow = W + (size_t)(n0 + l15) * K + 2 * half;

    v8f c = {};
#pragma unroll 8
    for (int kk = 0; kk < K; kk += 4) {
        v2f a = *(const v2f*)(xrow + kk);
        v2f b = *(const v2f*)(wrow + kk);
        // D = A(16x4,f32) x B(4x16,f32) + C(16x16,f32)
        c = __builtin_amdgcn_wmma_f32_16x16x4_f32(
                /*neg_a=*/false, a, /*neg_b=*/false, b,
                /*c_mod=*/(short)0, c, /*reuse_a=*/false, /*reuse_b=*/false);
    }

    const float bval = bias ? bias[n0 + l15] : 0.0f;
#pragma unroll
    for (int r = 0; r < 8; ++r) {
        const int row = m0 + r + 8 * half;
        float val = c[r] + bval;
        if (addsrc) val += addsrc[(size_t)row * N + n0 + l15];
        if (relu)   val = fmaxf(val, 0.0f);
        Y[(size_t)row * N + n0 + l15] = val;
    }
}

// Fused GRUCell (x = v-as-updates via gi, h = slots) + LayerNorm(h)*g+b.
// One block (256 threads) per slot row; each thread owns 3 of the 768 dims.
__global__ __launch_bounds__(256) void gru_ln_kernel(
    const float* __restrict__ gi, const float* __restrict__ gh,
    const float* __restrict__ slots, const float* __restrict__ g_ff,
    const float* __restrict__ beta_ff,
    float* __restrict__ h_out, float* __restrict__ pre_out)
{
    __shared__ float red[256];
    const int i = blockIdx.x;
    const int t = threadIdx.x;

    float hv[3];
    float lsum = 0.f;
#pragma unroll
    for (int e = 0; e < 3; ++e) {
        const int d  = t + 256 * e;
        const float ir = gi[i * THREE_D + d];
        const float iz = gi[i * THREE_D + 768 + d];
        const float in_ = gi[i * THREE_D + 1536 + d];
        const float hr = gh[i * THREE_D + d];
        const float hz = gh[i * THREE_D + 768 + d];
        const float hn = gh[i * THREE_D + 1536 + d];
        const float r = 1.f / (1.f + __expf(-(ir + hr)));
        const float z = 1.f / (1.f + __expf(-(iz + hz)));
        const float n = tanhf(in_ + r * hn);
        const float s = slots[i * D_DIM + d];
        const float h = (1.f - z) * n + z * s;
        hv[e] = h;
        h_out[i * D_DIM + d] = h;
        lsum += h;
    }

    red[t] = lsum; __syncthreads();
    for (int s = 128; s > 0; s >>= 1) { if (t < s) red[t] += red[t + s]; __syncthreads(); }
    const float mean = red[0] * (1.0f / 768.0f);
    __syncthreads();

    float lvar = 0.f;
#pragma unroll
    for (int e = 0; e < 3; ++e) { const float dlt = hv[e] - mean; lvar += dlt * dlt; }
    red[t] = lvar; __syncthreads();
    for (int s = 128; s > 0; s >>= 1) { if (t < s) red[t] += red[t + s]; __syncthreads(); }
    const float inv = rsqrtf(red[0] * (1.0f / 768.0f) + LN_EPS);

#pragma unroll
    for (int e = 0; e < 3; ++e) {
        const int d = t + 256 * e;
        pre_out[i * D_DIM + d] = (hv[e] - mean) * inv * g_ff[d] + beta_ff[d];
    }
}

extern "C" void kernel_launch(void* const* d_in, const int* in_sizes, int n_in,
                              void* d_out, int out_size, void* d_ws, size_t ws_size,
                              hipStream_t stream) {
    (void)in_sizes; (void)n_in; (void)out_size; (void)ws_size;
    // setup_inputs() order:
    const float* cand    = (const float*)d_in[0];
    // d_in[1..5]: pano_feat, Wq, bq, Wk, bk -- provably unused (softmax identity)
    const float* Wv      = (const float*)d_in[6];
    const float* bv      = (const float*)d_in[7];
    const float* W_ih    = (const float*)d_in[8];
    const float* b_ih    = (const float*)d_in[9];
    const float* W_hh    = (const float*)d_in[10];
    const float* b_hh    = (const float*)d_in[11];
    const float* W1      = (const float*)d_in[12];
    const float* b1      = (const float*)d_in[13];
    const float* W2      = (const float*)d_in[14];
    const float* b2      = (const float*)d_in[15];
    const float* g_ff    = (const float*)d_in[16];
    const float* beta_ff = (const float*)d_in[17];
    // d_in[18..19]: g_in, beta_in -- unused (input LN feeds only q)

    float* ws    = (float*)d_ws;
    float* v_buf = ws;                   // 128*768   = 98304
    float* gi    = v_buf + 98304;        // 128*2304  = 294912
    float* gh    = gi    + 294912;       // 128*2304
    float* h     = gh    + 294912;       // 128*768
    float* pre   = h     + 98304;        // 128*768
    float* ff1   = pre   + 98304;        // 128*3072  = 393216
    float* slots = ff1   + 393216;       // 128*768
    float* out   = (float*)d_out;

    const dim3 blk(256);

    // v  = cand @ Wv^T + bv                      [128,768]
    gemm_xwt_wmma<<<D_DIM / 16, blk, 0, stream>>>(cand, Wv, bv, nullptr, v_buf,
                                                  D_DIM, D_DIM, 0);
    // gi = v @ W_ih^T + b_ih  (loop-invariant)   [128,2304]
    gemm_xwt_wmma<<<THREE_D / 16, blk, 0, stream>>>(v_buf, W_ih, b_ih, nullptr, gi,
                                                    D_DIM, THREE_D, 0);

    const float* cur = cand;  // slots_0 = cand_feat
    for (int it = 0; it < 3; ++it) {
        // gh = slots @ W_hh^T + b_hh             [128,2304]
        gemm_xwt_wmma<<<THREE_D / 16, blk, 0, stream>>>(cur, W_hh, b_hh, nullptr, gh,
                                                        D_DIM, THREE_D, 0);
        // GRU gates + h, then pre = LN(h)*g_ff + beta_ff
        gru_ln_kernel<<<NSLOTS, blk, 0, stream>>>(gi, gh, cur, g_ff, beta_ff, h, pre);
        // ff1 = relu(pre @ W1^T + b1)            [128,3072]
        gemm_xwt_wmma<<<H_DIM / 16, blk, 0, stream>>>(pre, W1, b1, nullptr, ff1,
                                                      D_DIM, H_DIM, 1);
        // slots_{t+1} = h + ff1 @ W2^T + b2      [128,768]  (last iter -> d_out)
        float* dst = (it == 2) ? out : slots;
        gemm_xwt_wmma<<<D_DIM / 16, blk, 0, stream>>>(ff1, W2, b2, h, dst,
                                                      H_DIM, D_DIM, 0);
        cur = dst;
    }
}